// GATv2_34849364639936
// MI455X (gfx1250) — compile-verified
//
#include <hip/hip_runtime.h>

typedef __attribute__((ext_vector_type(2))) float v2f;
typedef __attribute__((ext_vector_type(8))) float v8f;

#define NEG_SLOPE 0.2f

// monotone uint key for float atomic-max (order preserving, -0 < +0)
__device__ __forceinline__ unsigned fkey(float f) {
  unsigned u = __float_as_uint(f);
  return (u & 0x80000000u) ? ~u : (u | 0x80000000u);
}
__device__ __forceinline__ float funkey(unsigned k) {
  unsigned u = (k & 0x80000000u) ? (k ^ 0x80000000u) : ~k;
  return __uint_as_float(u);
}

__device__ __forceinline__ float lrelu(float t) { return t > 0.0f ? t : NEG_SLOPE * t; }

__device__ __forceinline__ void edge_sd(const long long* __restrict__ ei,
                                        long long E0, long long e, int& s, int& d) {
  if (e < E0) { s = (int)ei[e]; d = (int)ei[E0 + e]; }
  else        { s = d = (int)(e - E0); }          // self loops appended
}

// ---------------------------------------------------------- layer-1 WMMA GEMM
// xl = x @ W1l, xr = x @ W1r  (M=50000, K=128, N=64 each), exact fp32 via
// V_WMMA_F32_16X16X4_F32.  One wave per 16-row M-tile, 4 waves/block.
// W1l/W1r staged TRANSPOSED in LDS ([col][k]) so B fragments are b64 loads.
__global__ void __launch_bounds__(128)
k_gemm_wmma(const float* __restrict__ x,
            const float* __restrict__ Wl, const float* __restrict__ Wr,
            float* __restrict__ xl, float* __restrict__ xr, int nNodes) {
  __shared__ __align__(16) float sWlT[64 * 128];   // [col][k], 32 KB
  __shared__ __align__(16) float sWrT[64 * 128];   // [col][k], 32 KB
  int tid = threadIdx.x;
  for (int i = tid; i < 128 * 64; i += 128) {
    int k = i >> 6, col = i & 63;                  // W is [k][col], k<128, col<64
    sWlT[col * 128 + k] = Wl[i];
    sWrT[col * 128 + k] = Wr[i];
  }
  __syncthreads();

  int wave = tid >> 5, lane = tid & 31;
  int tile = blockIdx.x * 4 + wave;
  int row0 = tile * 16;
  if (row0 >= nNodes) return;

  int m     = lane & 15;          // A row within tile
  int koff  = (lane & 16) >> 3;   // K offset: 0 (lanes 0-15) or 2 (lanes 16-31)
  int n     = lane & 15;          // B/C/D column
  int mbase = (lane >> 4) << 3;   // C/D row offset: 0 or 8

  v8f acc[8] = {};                // 4 N-tiles for Wl, 4 for Wr

  const float* arowBase = x + (size_t)(row0 + m) * 128 + koff;

#pragma unroll 4
  for (int kc = 0; kc < 32; ++kc) {               // K in steps of 4
    // A fragment 16x4 f32: VGPR0 = K=koff, VGPR1 = K=koff+1  -> b64 load
    v2f a = *(const v2f*)(arowBase + kc * 4);

#pragma unroll
    for (int nt = 0; nt < 4; ++nt) {
      // B fragment 4x16 f32: lane = column, same K mapping as A -> b64 loads
      const float* pl = &sWlT[(nt * 16 + n) * 128 + kc * 4 + koff];
      const float* pr = &sWrT[(nt * 16 + n) * 128 + kc * 4 + koff];
      v2f bl = *(const v2f*)(pl);
      v2f br = *(const v2f*)(pr);
      acc[nt]     = __builtin_amdgcn_wmma_f32_16x16x4_f32(
          false, a, false, bl, (short)0, acc[nt], false, false);
      acc[4 + nt] = __builtin_amdgcn_wmma_f32_16x16x4_f32(
          false, a, false, br, (short)0, acc[4 + nt], false, false);
    }
  }

#pragma unroll
  for (int nt = 0; nt < 4; ++nt) {
#pragma unroll
    for (int v = 0; v < 8; ++v) {
      int row = row0 + mbase + v;
      xl[(size_t)row * 64 + nt * 16 + n] = acc[nt][v];
      xr[(size_t)row * 64 + nt * 16 + n] = acc[4 + nt][v];
    }
  }
}

// ----------------------------------------------------- layer-1 edge softmax
// thread = (edge, head)
__global__ void k_score1(const long long* __restrict__ ei, long long E0, long long Etot,
                         const float* __restrict__ xl, const float* __restrict__ xr,
                         const float* __restrict__ att,
                         float* __restrict__ score, unsigned* __restrict__ mkey) {
  long long idx = (long long)blockIdx.x * blockDim.x + threadIdx.x;
  if (idx >= Etot * 8) return;
  long long e = idx >> 3;
  int h = (int)(idx & 7);
  int s, d; edge_sd(ei, E0, e, s, d);
  const float4* pl = (const float4*)(xl + (size_t)s * 64 + h * 8);
  const float4* pr = (const float4*)(xr + (size_t)d * 64 + h * 8);
  float4 l0 = pl[0], l1 = pl[1], r0 = pr[0], r1 = pr[1];
  const float* a = att + h * 8;
  float sc = a[0] * lrelu(l0.x + r0.x) + a[1] * lrelu(l0.y + r0.y) +
             a[2] * lrelu(l0.z + r0.z) + a[3] * lrelu(l0.w + r0.w) +
             a[4] * lrelu(l1.x + r1.x) + a[5] * lrelu(l1.y + r1.y) +
             a[6] * lrelu(l1.z + r1.z) + a[7] * lrelu(l1.w + r1.w);
  score[idx] = sc;
  atomicMax(&mkey[(size_t)d * 8 + h], fkey(sc));
}

__global__ void k_expsum1(const long long* __restrict__ ei, long long E0, long long Etot,
                          const unsigned* __restrict__ mkey,
                          float* __restrict__ score, float* __restrict__ denom) {
  long long idx = (long long)blockIdx.x * blockDim.x + threadIdx.x;
  if (idx >= Etot * 8) return;
  long long e = idx >> 3;
  int h = (int)(idx & 7);
  int s, d; edge_sd(ei, E0, e, s, d);
  float num = __expf(score[idx] - funkey(mkey[(size_t)d * 8 + h]));
  score[idx] = num;
  atomicAdd(&denom[(size_t)d * 8 + h], num);
}

__global__ void k_agg1(const long long* __restrict__ ei, long long E0, long long Etot,
                       const float* __restrict__ score, const float* __restrict__ denom,
                       const float* __restrict__ xl, float* __restrict__ agg) {
  long long idx = (long long)blockIdx.x * blockDim.x + threadIdx.x;
  if (idx >= Etot * 8) return;
  long long e = idx >> 3;
  int h = (int)(idx & 7);
  int s, d; edge_sd(ei, E0, e, s, d);
  float alpha = score[idx] / denom[(size_t)d * 8 + h];
  const float4* pl = (const float4*)(xl + (size_t)s * 64 + h * 8);
  float4 l0 = pl[0], l1 = pl[1];
  float* o = agg + (size_t)d * 64 + h * 8;
  atomicAdd(o + 0, alpha * l0.x); atomicAdd(o + 1, alpha * l0.y);
  atomicAdd(o + 2, alpha * l0.z); atomicAdd(o + 3, alpha * l0.w);
  atomicAdd(o + 4, alpha * l1.x); atomicAdd(o + 5, alpha * l1.y);
  atomicAdd(o + 6, alpha * l1.z); atomicAdd(o + 7, alpha * l1.w);
}

// -------------------------------------- ELU + fused layer-2 projections (64->1)
__global__ void k_elu_proj(const float* __restrict__ agg, const float* __restrict__ b1,
                           const float* __restrict__ W2l, const float* __restrict__ W2r,
                           float* __restrict__ hl, float* __restrict__ hr, int nNodes) {
  int nIdx = blockIdx.x * blockDim.x + threadIdx.x;
  if (nIdx >= nNodes) return;
  const float* row = agg + (size_t)nIdx * 64;
  float dl = 0.0f, dr = 0.0f;
#pragma unroll
  for (int c = 0; c < 64; ++c) {
    float v = row[c] + b1[c];
    v = v > 0.0f ? v : (__expf(v) - 1.0f);       // ELU
    dl += v * W2l[c];
    dr += v * W2r[c];
  }
  hl[nIdx] = dl;
  hr[nIdx] = dr;
}

// ----------------------------------------------------- layer-2 edge softmax
__global__ void k_score2(const long long* __restrict__ ei, long long E0, long long Etot,
                         const float* __restrict__ hl, const float* __restrict__ hr,
                         const float* __restrict__ att2,
                         float* __restrict__ score, unsigned* __restrict__ mkey) {
  long long e = (long long)blockIdx.x * blockDim.x + threadIdx.x;
  if (e >= Etot) return;
  int s, d; edge_sd(ei, E0, e, s, d);
  float sc = att2[0] * lrelu(hl[s] + hr[d]);
  score[e] = sc;
  atomicMax(&mkey[d], fkey(sc));
}

__global__ void k_expsum2(const long long* __restrict__ ei, long long E0, long long Etot,
                          const unsigned* __restrict__ mkey,
                          float* __restrict__ score, float* __restrict__ denom) {
  long long e = (long long)blockIdx.x * blockDim.x + threadIdx.x;
  if (e >= Etot) return;
  int s, d; edge_sd(ei, E0, e, s, d);
  float num = __expf(score[e] - funkey(mkey[d]));
  score[e] = num;
  atomicAdd(&denom[d], num);
}

__global__ void k_agg2(const long long* __restrict__ ei, long long E0, long long Etot,
                       const float* __restrict__ score, const float* __restrict__ denom,
                       const float* __restrict__ hl, float* __restrict__ outagg) {
  long long e = (long long)blockIdx.x * blockDim.x + threadIdx.x;
  if (e >= Etot) return;
  int s, d; edge_sd(ei, E0, e, s, d);
  atomicAdd(&outagg[d], (score[e] / denom[d]) * hl[s]);
}

__global__ void k_finalize(const float* __restrict__ outagg, const float* __restrict__ b2,
                           float* __restrict__ out, int nNodes) {
  int nIdx = blockIdx.x * blockDim.x + threadIdx.x;
  if (nIdx < nNodes) out[nIdx] = outagg[nIdx] + b2[0];
}

// ---------------------------------------------------------------------------
extern "C" void kernel_launch(void* const* d_in, const int* in_sizes, int n_in,
                              void* d_out, int out_size, void* d_ws, size_t ws_size,
                              hipStream_t stream) {
  const float*     x    = (const float*)d_in[0];
  const long long* ei   = (const long long*)d_in[1];   // int64 edge_index [2,E0]
  const float*     W1l  = (const float*)d_in[2];
  const float*     W1r  = (const float*)d_in[3];
  const float*     att1 = (const float*)d_in[4];
  const float*     b1   = (const float*)d_in[5];
  const float*     W2l  = (const float*)d_in[6];
  const float*     W2r  = (const float*)d_in[7];
  const float*     att2 = (const float*)d_in[8];
  const float*     b2   = (const float*)d_in[9];

  const long long nNodes = in_sizes[0] / 128;   // 50000
  const long long E0     = in_sizes[1] / 2;     // 1600000
  const long long Etot   = E0 + nNodes;         // + self loops

  // workspace carve-out (256B aligned)
  char* cur = (char*)d_ws;
  auto alloc = [&](size_t bytes) -> void* {
    void* p = (void*)cur;
    cur += (bytes + 255) & ~(size_t)255;
    return p;
  };
  float*    xl     = (float*)   alloc((size_t)nNodes * 64 * sizeof(float));
  float*    xr     = (float*)   alloc((size_t)nNodes * 64 * sizeof(float));
  float*    score1 = (float*)   alloc((size_t)Etot * 8 * sizeof(float));
  unsigned* mkey1  = (unsigned*)alloc((size_t)nNodes * 8 * sizeof(unsigned));
  float*    denom1 = (float*)   alloc((size_t)nNodes * 8 * sizeof(float));
  float*    agg1   = (float*)   alloc((size_t)nNodes * 64 * sizeof(float));
  float*    hl     = (float*)   alloc((size_t)nNodes * sizeof(float));
  float*    hr     = (float*)   alloc((size_t)nNodes * sizeof(float));
  float*    score2 = (float*)   alloc((size_t)Etot * sizeof(float));
  unsigned* mkey2  = (unsigned*)alloc((size_t)nNodes * sizeof(unsigned));
  float*    denom2 = (float*)   alloc((size_t)nNodes * sizeof(float));
  float*    outagg = (float*)   alloc((size_t)nNodes * sizeof(float));

  // zero accumulators (mkey=0 == key below fkey(-inf); every node has a self loop)
  hipMemsetAsync(mkey1,  0, (size_t)nNodes * 8 * sizeof(unsigned), stream);
  hipMemsetAsync(denom1, 0, (size_t)nNodes * 8 * sizeof(float),    stream);
  hipMemsetAsync(agg1,   0, (size_t)nNodes * 64 * sizeof(float),   stream);
  hipMemsetAsync(mkey2,  0, (size_t)nNodes * sizeof(unsigned),     stream);
  hipMemsetAsync(denom2, 0, (size_t)nNodes * sizeof(float),        stream);
  hipMemsetAsync(outagg, 0, (size_t)nNodes * sizeof(float),        stream);

  const int TB = 256;

  int mTiles = (int)((nNodes + 15) / 16);               // 3125
  int gemmBlocks = (mTiles + 3) / 4;                    // 4 waves/block
  k_gemm_wmma<<<gemmBlocks, 128, 0, stream>>>(x, W1l, W1r, xl, xr, (int)nNodes);

  long long eh = Etot * 8;
  unsigned gEH = (unsigned)((eh + TB - 1) / TB);
  unsigned gE  = (unsigned)((Etot + TB - 1) / TB);
  unsigned gN  = (unsigned)((nNodes + TB - 1) / TB);

  k_score1 <<<gEH, TB, 0, stream>>>(ei, E0, Etot, xl, xr, att1, score1, mkey1);
  k_expsum1<<<gEH, TB, 0, stream>>>(ei, E0, Etot, mkey1, score1, denom1);
  k_agg1   <<<gEH, TB, 0, stream>>>(ei, E0, Etot, score1, denom1, xl, agg1);

  k_elu_proj<<<gN, TB, 0, stream>>>(agg1, b1, W2l, W2r, hl, hr, (int)nNodes);

  k_score2 <<<gE, TB, 0, stream>>>(ei, E0, Etot, hl, hr, att2, score2, mkey2);
  k_expsum2<<<gE, TB, 0, stream>>>(ei, E0, Etot, mkey2, score2, denom2);
  k_agg2   <<<gE, TB, 0, stream>>>(ei, E0, Etot, score2, denom2, hl, outagg);

  k_finalize<<<gN, TB, 0, stream>>>(outagg, b2, (float*)d_out, (int)nNodes);
}